// GraphAutoencoder_86466281603448
// MI455X (gfx1250) — compile-verified
//
#include <hip/hip_runtime.h>
#include <math.h>

// ---------------------------------------------------------------------------
// GraphAutoencoder for MI455X (gfx1250, wave32, WMMA).
// All matmuls run on V_WMMA_F32_16X16X4_F32 (f32 matrix pipe, keeps reference
// precision). LSTM1 stages its 192KB of weights into LDS via the Tensor Data
// Mover (TENSOR_LOAD_TO_LDS + s_wait_tensorcnt). LSTM2 is a 20000-step serial
// chain -> single persistent workgroup with register-resident weights.
// ---------------------------------------------------------------------------

typedef float v2f __attribute__((ext_vector_type(2)));
typedef float v8f __attribute__((ext_vector_type(8)));
typedef unsigned int v4u __attribute__((ext_vector_type(4)));
typedef int v4i __attribute__((ext_vector_type(4)));
typedef int v8i __attribute__((ext_vector_type(8)));

#define DEVFN __device__ __forceinline__

constexpr int N_NODES = 20000;
constexpr int T_STEPS = 16;
constexpr int F_DIM   = 64;
constexpr int F_INCH  = 65;
constexpr int HID     = 128;
constexpr int LAT     = 64;
constexpr int E_EDGES = 320000;

constexpr size_t TNH = (size_t)T_STEPS * N_NODES * HID;   // 40,960,000 floats
constexpr size_t TNF = (size_t)T_STEPS * N_NODES * F_DIM; // 20,480,000 floats

DEVFN v8f wmma4(v2f a, v2f b, v8f c) {
  // D(16x16,f32) = A(16x4,f32) * B(4x16,f32) + C
  return __builtin_amdgcn_wmma_f32_16x16x4_f32(false, a, false, b,
                                               (short)0, c, false, false);
}
DEVFN float sigmoidf(float x) { return 1.0f / (1.0f + expf(-x)); }

// ---------------------------------------------------------------------------
// TDM: DMA a dense row-major (rows x cols) f32 matrix from global memory into
// LDS at byte offset lds_off. D# layout per CDNA5 ISA ch.8 (CDNA5 ISA):
//   group0: count=1 | lds_addr[63:32] | global_addr[120:64] | type=2[127:126]
//   group1: data_size=4B | tensor_dim0=cols | tensor_dim1=rows |
//           tile_dim0=cols | tile_dim1=rows | dim0_stride=cols
//   remaining groups: zero (2-D tensor)
// Uses the 6-arg clang-23 builtin form: (v4u, v8i, v4i, v4i, v8i, i32 cpol).
// ---------------------------------------------------------------------------
DEVFN void tdm_load_2d_f32(const void* gptr, unsigned lds_off,
                           unsigned rows, unsigned cols) {
  unsigned long long ga = (unsigned long long)gptr;
  v4u g0;
  g0.x = 1u;                                            // count = 1 descriptor
  g0.y = lds_off;                                       // lds_addr (bytes)
  g0.z = (unsigned)ga;                                  // global_addr[31:0]
  g0.w = (unsigned)((ga >> 32) & 0x01FFFFFFu) | 0x80000000u; // [56:32] | type=2
  v8i g1;
  g1[0] = (int)0x00020000u;                             // data_size = 2 (4B)
  g1[1] = (int)((cols & 0xFFFFu) << 16);                // tensor_dim0[15:0]
  g1[2] = (int)(((cols >> 16) & 0xFFFFu) | ((rows & 0xFFFFu) << 16));
  g1[3] = (int)(((rows >> 16) & 0xFFFFu) | ((cols & 0xFFFFu) << 16)); // tile0
  g1[4] = (int)(rows & 0xFFFFu);                        // tile_dim1 (tile2=0)
  g1[5] = (int)cols;                                    // dim0_stride[31:0]
  g1[6] = 0;
  g1[7] = 0;
  v4i z4 = {0, 0, 0, 0};
  v8i z8 = {0, 0, 0, 0, 0, 0, 0, 0};
  __builtin_amdgcn_tensor_load_to_lds(g0, g1, z4, z4, z8, 0);
}

// ---------------------------------------------------------------------------
// Degree / dinv
// ---------------------------------------------------------------------------
__global__ void k_deg(const long long* __restrict__ ei, float* __restrict__ deg) {
  int e = blockIdx.x * blockDim.x + threadIdx.x;
  if (e < E_EDGES) atomicAdd(&deg[(int)ei[E_EDGES + e]], 1.0f);
}

__global__ void k_dinv(float* __restrict__ d) {
  int n = blockIdx.x * blockDim.x + threadIdx.x;
  if (n < N_NODES) d[n] = rsqrtf(d[n] + 1.0f);
}

// ---------------------------------------------------------------------------
// Encoder matmul: xw[t,n,h] = (x[n,t,:64]*x[n,t,64]) @ enc_W   (K=64 -> 128)
// block = 256 threads = 8 waves, one 16x16 col tile per wave. grid (1250, 16)
// ---------------------------------------------------------------------------
__global__ void k_enc_mm(const float* __restrict__ x, const float* __restrict__ W,
                         float* __restrict__ xw) {
  __shared__ float sX[16 * 64];
  __shared__ float sW[64 * 128];
  const int tile = blockIdx.x, t = blockIdx.y, tid = threadIdx.x;
  for (int i = tid; i < 16 * 64; i += 256) {
    int m = i >> 6, k = i & 63;
    const float* row = x + ((size_t)(tile * 16 + m) * T_STEPS + t) * F_INCH;
    sX[i] = row[k] * row[64];               // apply importance weighting
  }
  for (int i = tid; i < 64 * 128; i += 256) sW[i] = W[i];
  __syncthreads();

  const int w = tid >> 5, lane = tid & 31, half = lane >> 4, l15 = lane & 15;
  const int nb = w * 16;
  v8f acc = {0, 0, 0, 0, 0, 0, 0, 0};
#pragma unroll
  for (int kb = 0; kb < 16; ++kb) {
    int k0 = kb * 4 + half * 2;
    v2f a, b;
    a.x = sX[l15 * 64 + k0];       a.y = sX[l15 * 64 + k0 + 1];
    b.x = sW[k0 * 128 + nb + l15]; b.y = sW[(k0 + 1) * 128 + nb + l15];
    acc = wmma4(a, b, acc);
  }
#pragma unroll
  for (int r = 0; r < 8; ++r) {
    int m = r + 8 * half;
    xw[((size_t)t * N_NODES + tile * 16 + m) * HID + nb + l15] = acc[r];
  }
}

// ---------------------------------------------------------------------------
// GCN edge scatter-add: agg[t,dst,:] += xw[t,src,:] * dinv[src]*dinv[dst]
// one block per edge, blockDim == C channels, loops over T
// ---------------------------------------------------------------------------
__global__ void k_gcn_agg(const float* __restrict__ xw, const long long* __restrict__ ei,
                          const float* __restrict__ dinv, float* __restrict__ agg, int C) {
  int e = blockIdx.x, h = threadIdx.x;
  int s = (int)ei[e], d = (int)ei[E_EDGES + e];
  float nm = dinv[s] * dinv[d];
  for (int t = 0; t < T_STEPS; ++t)
    atomicAdd(&agg[((size_t)t * N_NODES + d) * C + h],
              xw[((size_t)t * N_NODES + s) * C + h] * nm);
}

// ---------------------------------------------------------------------------
// Encoder LN (in place on agg): v = agg + dinv^2*xw + enc_b ; LN(v)*g + b
// ---------------------------------------------------------------------------
__global__ void k_ln_enc(float* __restrict__ agg, const float* __restrict__ xw,
                         const float* __restrict__ dinv, const float* __restrict__ bias,
                         const float* __restrict__ g, const float* __restrict__ b) {
  __shared__ float s1[128], s2[128];
  int n = blockIdx.x, t = blockIdx.y, h = threadIdx.x;
  size_t idx = ((size_t)t * N_NODES + n) * HID + h;
  float di = dinv[n];
  float v = agg[idx] + di * di * xw[idx] + bias[h];
  s1[h] = v; s2[h] = v * v;
  __syncthreads();
  for (int s = 64; s > 0; s >>= 1) {
    if (h < s) { s1[h] += s1[h + s]; s2[h] += s2[h + s]; }
    __syncthreads();
  }
  float m = s1[0] * (1.0f / 128.0f);
  float var = s2[0] * (1.0f / 128.0f) - m * m;
  agg[idx] = (v - m) * rsqrtf(var + 1e-5f) * g[h] + b[h];
}

// ---------------------------------------------------------------------------
// Decoder LN (in place): LN(hd)*g + b
// ---------------------------------------------------------------------------
__global__ void k_ln_dec(float* __restrict__ hd, const float* __restrict__ g,
                         const float* __restrict__ b) {
  __shared__ float s1[128], s2[128];
  int n = blockIdx.x, t = blockIdx.y, h = threadIdx.x;
  size_t idx = ((size_t)t * N_NODES + n) * HID + h;
  float v = hd[idx];
  s1[h] = v; s2[h] = v * v;
  __syncthreads();
  for (int s = 64; s > 0; s >>= 1) {
    if (h < s) { s1[h] += s1[h + s]; s2[h] += s2[h + s]; }
    __syncthreads();
  }
  float m = s1[0] * (1.0f / 128.0f);
  float var = s2[0] * (1.0f / 128.0f) - m * m;
  hd[idx] = (v - m) * rsqrtf(var + 1e-5f) * g[h] + b[h];
}

// ---------------------------------------------------------------------------
// LSTM1: parallel over 16-node tiles. 512 threads = 16 waves, one 16-col gate
// tile per wave (4*LAT = 256 cols). Both weight matrices DMAed into LDS by
// the Tensor Data Mover (wave 0 issues two descriptors, waits TENSORcnt==0,
// workgroup barrier publishes). Recurrence over T=16.
// dyn LDS: Wih(32768) Whh(16384) X(2048) H(1024) G(4096) floats = 225,280 B
// ---------------------------------------------------------------------------
__global__ void k_lstm1(const float* __restrict__ hin, const float* __restrict__ Wih,
                        const float* __restrict__ Whh, const float* __restrict__ bvec,
                        float* __restrict__ zout) {
  extern __shared__ float sm1[];
  float* sWih = sm1;                  // 128*256   @ LDS byte offset 0
  float* sWhh = sWih + 128 * 256;     // 64*256    @ LDS byte offset 131072
  float* sX   = sWhh + 64 * 256;      // 16*128
  float* sH   = sX + 16 * 128;        // 16*64
  float* sG   = sH + 16 * 64;         // 16*256
  const int tid = threadIdx.x, tile = blockIdx.x;

  if (tid < 32) {                     // wave 0: TDM weight staging
    tdm_load_2d_f32(Wih, 0u,      128u, 256u);
    tdm_load_2d_f32(Whh, 131072u,  64u, 256u);
    __builtin_amdgcn_s_wait_tensorcnt(0);
  }
  for (int i = tid; i < 16 * 64; i += 512) sH[i] = 0.0f;

  const int w = tid >> 5, lane = tid & 31, half = lane >> 4, l15 = lane & 15;
  const int nb = w * 16;
  const float bias = bvec[nb + l15];
  float c0 = 0.0f, c1 = 0.0f;
  __syncthreads();

  for (int t = 0; t < T_STEPS; ++t) {
    for (int i = tid; i < 16 * 128; i += 512) {
      int m = i >> 7, k = i & 127;
      sX[i] = hin[((size_t)t * N_NODES + tile * 16 + m) * HID + k];
    }
    __syncthreads();

    v8f acc = {bias, bias, bias, bias, bias, bias, bias, bias};
#pragma unroll
    for (int kb = 0; kb < 32; ++kb) {            // xt @ Wih  (K = 128)
      int k0 = kb * 4 + half * 2;
      v2f a, b;
      a.x = sX[l15 * 128 + k0];       a.y = sX[l15 * 128 + k0 + 1];
      b.x = sWih[k0 * 256 + nb + l15]; b.y = sWih[(k0 + 1) * 256 + nb + l15];
      acc = wmma4(a, b, acc);
    }
#pragma unroll
    for (int kb = 0; kb < 16; ++kb) {            // h @ Whh  (K = 64)
      int k0 = kb * 4 + half * 2;
      v2f a, b;
      a.x = sH[l15 * 64 + k0];        a.y = sH[l15 * 64 + k0 + 1];
      b.x = sWhh[k0 * 256 + nb + l15]; b.y = sWhh[(k0 + 1) * 256 + nb + l15];
      acc = wmma4(a, b, acc);
    }
#pragma unroll
    for (int r = 0; r < 8; ++r)
      sG[(r + 8 * half) * 256 + nb + l15] = acc[r];
    __syncthreads();

    {   // 1024 (m,l) cells; this thread owns tid and tid+512
      int m = tid >> 6, l = tid & 63;
      float gi = sG[m * 256 + l],        gf = sG[m * 256 + 64 + l];
      float gg = sG[m * 256 + 128 + l],  go = sG[m * 256 + 192 + l];
      c0 = sigmoidf(gf) * c0 + sigmoidf(gi) * tanhf(gg);
      float h = sigmoidf(go) * tanhf(c0);
      sH[tid] = h;
      zout[(size_t)(tile * 16 + m) * (T_STEPS * LAT) + t * LAT + l] = h;
    }
    {
      int idx = tid + 512;
      int m = idx >> 6, l = idx & 63;
      float gi = sG[m * 256 + l],        gf = sG[m * 256 + 64 + l];
      float gg = sG[m * 256 + 128 + l],  go = sG[m * 256 + 192 + l];
      c1 = sigmoidf(gf) * c1 + sigmoidf(gi) * tanhf(gg);
      float h = sigmoidf(go) * tanhf(c1);
      sH[idx] = h;
      zout[(size_t)(tile * 16 + m) * (T_STEPS * LAT) + t * LAT + l] = h;
    }
    __syncthreads();
  }
}

// ---------------------------------------------------------------------------
// LSTM2: 20000 sequential steps, batch 16 (the reference scans z_nt over N).
// Single persistent workgroup: 1024 threads = 32 waves, each wave owns one
// 16-wide gate-column tile of the 512 gate columns. Weights (393KB) exceed
// LDS, so B-fragments are REGISTER-RESIDENT per wave (96 VGPRs/lane), loaded
// once. z row staged in LDS each step + prefetch of next row.
// dyn LDS: Z(1024) H(2048) G(8192) floats = 45,056 B
// ---------------------------------------------------------------------------
__global__ void k_lstm2(const float* __restrict__ zsrc, const float* __restrict__ Wih,
                        const float* __restrict__ Whh, const float* __restrict__ bvec,
                        float* __restrict__ hd) {
  extern __shared__ float sm2[];
  float* sZ = sm2;            // 16*64
  float* sH = sZ + 1024;      // 16*128
  float* sG = sH + 2048;      // 16*512
  const int tid = threadIdx.x;
  const int w = tid >> 5, lane = tid & 31, half = lane >> 4, l15 = lane & 15;
  const int ncol = w * 16 + l15;

  v2f bih[16];                                   // Wih2 64x512 slice
#pragma unroll
  for (int kb = 0; kb < 16; ++kb) {
    int k0 = kb * 4 + half * 2;
    bih[kb].x = Wih[(size_t)k0 * 512 + ncol];
    bih[kb].y = Wih[(size_t)(k0 + 1) * 512 + ncol];
  }
  v2f bhh[32];                                   // Whh2 128x512 slice
#pragma unroll
  for (int kb = 0; kb < 32; ++kb) {
    int k0 = kb * 4 + half * 2;
    bhh[kb].x = Whh[(size_t)k0 * 512 + ncol];
    bhh[kb].y = Whh[(size_t)(k0 + 1) * 512 + ncol];
  }
  const float bias = bvec[ncol];

  for (int i = tid; i < 2048; i += 1024) sH[i] = 0.0f;
  float c0 = 0.0f, c1 = 0.0f;
  __syncthreads();

  for (int n = 0; n < N_NODES; ++n) {
    sZ[tid] = zsrc[(size_t)n * 1024 + tid];      // z_nt[n] : (16,64)
    if (n + 1 < N_NODES)
      __builtin_prefetch(zsrc + (size_t)(n + 1) * 1024 + tid, 0, 1);
    __syncthreads();

    v8f acc = {bias, bias, bias, bias, bias, bias, bias, bias};
#pragma unroll
    for (int kb = 0; kb < 16; ++kb) {            // xt @ Wih2 (K = 64)
      int k0 = kb * 4 + half * 2;
      v2f a;
      a.x = sZ[l15 * 64 + k0]; a.y = sZ[l15 * 64 + k0 + 1];
      acc = wmma4(a, bih[kb], acc);
    }
#pragma unroll
    for (int kb = 0; kb < 32; ++kb) {            // h @ Whh2 (K = 128)
      int k0 = kb * 4 + half * 2;
      v2f a;
      a.x = sH[l15 * 128 + k0]; a.y = sH[l15 * 128 + k0 + 1];
      acc = wmma4(a, bhh[kb], acc);
    }
#pragma unroll
    for (int r = 0; r < 8; ++r)
      sG[(r + 8 * half) * 512 + ncol] = acc[r];
    __syncthreads();

    {   // 2048 (t,hc) cells; thread owns tid and tid+1024
      int tt = tid >> 7, hc = tid & 127;
      float gi = sG[tt * 512 + hc],       gf = sG[tt * 512 + 128 + hc];
      float gg = sG[tt * 512 + 256 + hc], go = sG[tt * 512 + 384 + hc];
      c0 = sigmoidf(gf) * c0 + sigmoidf(gi) * tanhf(gg);
      float h = sigmoidf(go) * tanhf(c0);
      sH[tid] = h;
      hd[((size_t)tt * N_NODES + n) * HID + hc] = h;   // hd in (T,N,HID)
    }
    {
      int idx = tid + 1024;
      int tt = idx >> 7, hc = idx & 127;
      float gi = sG[tt * 512 + hc],       gf = sG[tt * 512 + 128 + hc];
      float gg = sG[tt * 512 + 256 + hc], go = sG[tt * 512 + 384 + hc];
      c1 = sigmoidf(gf) * c1 + sigmoidf(gi) * tanhf(gg);
      float h = sigmoidf(go) * tanhf(c1);
      sH[idx] = h;
      hd[((size_t)tt * N_NODES + n) * HID + hc] = h;
    }
    __syncthreads();
  }
}

// ---------------------------------------------------------------------------
// Decoder matmul: xw_dec[t,n,f] = hd_ln[t,n,:] @ dec_W   (K=128 -> 64)
// block = 128 threads = 4 waves (64 cols). grid (1250, 16)
// ---------------------------------------------------------------------------
__global__ void k_dec_mm(const float* __restrict__ hin, const float* __restrict__ W,
                         float* __restrict__ xw) {
  __shared__ float sX[16 * 128];
  __shared__ float sW[128 * 64];
  const int tile = blockIdx.x, t = blockIdx.y, tid = threadIdx.x;
  for (int i = tid; i < 16 * 128; i += 128) {
    int m = i >> 7, k = i & 127;
    sX[i] = hin[((size_t)t * N_NODES + tile * 16 + m) * HID + k];
  }
  for (int i = tid; i < 128 * 64; i += 128) sW[i] = W[i];
  __syncthreads();

  const int w = tid >> 5, lane = tid & 31, half = lane >> 4, l15 = lane & 15;
  const int nb = w * 16;
  v8f acc = {0, 0, 0, 0, 0, 0, 0, 0};
#pragma unroll
  for (int kb = 0; kb < 32; ++kb) {
    int k0 = kb * 4 + half * 2;
    v2f a, b;
    a.x = sX[l15 * 128 + k0];      a.y = sX[l15 * 128 + k0 + 1];
    b.x = sW[k0 * 64 + nb + l15];  b.y = sW[(k0 + 1) * 64 + nb + l15];
    acc = wmma4(a, b, acc);
  }
#pragma unroll
  for (int r = 0; r < 8; ++r)
    xw[((size_t)t * N_NODES + tile * 16 + r + 8 * half) * F_DIM + nb + l15] = acc[r];
}

// ---------------------------------------------------------------------------
// Final: reconstructed[n,t,f<64] = agg + dinv^2*xw + dec_b ; [n,t,64] = imp
// blockDim = 65 threads. grid (N, T)
// ---------------------------------------------------------------------------
__global__ void k_final(const float* __restrict__ agg, const float* __restrict__ xw,
                        const float* __restrict__ dinv, const float* __restrict__ bias,
                        const float* __restrict__ x, float* __restrict__ rec) {
  int n = blockIdx.x, t = blockIdx.y, f = threadIdx.x;
  size_t o = ((size_t)n * T_STEPS + t) * F_INCH + f;
  if (f < 64) {
    size_t idx = ((size_t)t * N_NODES + n) * F_DIM + f;
    float di = dinv[n];
    rec[o] = agg[idx] + di * di * xw[idx] + bias[f];
  } else {
    rec[o] = x[((size_t)n * T_STEPS + t) * F_INCH + 64];
  }
}

// ---------------------------------------------------------------------------
extern "C" void kernel_launch(void* const* d_in, const int* in_sizes, int n_in,
                              void* d_out, int out_size, void* d_ws, size_t ws_size,
                              hipStream_t stream) {
  const float* x        = (const float*)d_in[0];
  const float* enc_W    = (const float*)d_in[1];
  const float* enc_b    = (const float*)d_in[2];
  const float* enc_ln_g = (const float*)d_in[3];
  const float* enc_ln_b = (const float*)d_in[4];
  const float* l1_Wih   = (const float*)d_in[5];
  const float* l1_Whh   = (const float*)d_in[6];
  const float* l1_b     = (const float*)d_in[7];
  const float* l2_Wih   = (const float*)d_in[8];
  const float* l2_Whh   = (const float*)d_in[9];
  const float* l2_b     = (const float*)d_in[10];
  const float* dec_ln_g = (const float*)d_in[11];
  const float* dec_ln_b = (const float*)d_in[12];
  const float* dec_W    = (const float*)d_in[13];
  const float* dec_b    = (const float*)d_in[14];
  const long long* edge = (const long long*)d_in[15];

  float* ws   = (float*)d_ws;
  float* dinv = ws;                       // N (padded to 20480)
  float* bufA = ws + 20480;               // TNH floats: enc xw / dec xw+agg
  float* bufB = bufA + TNH;               // TNH floats: enc agg->h_enc -> hd

  float* zout = (float*)d_out;            // z_nt: (N,T,LAT) = TNF floats
  float* rec  = zout + TNF;               // reconstructed: (N,T,65)

  // --- degree / dinv ---
  (void)hipMemsetAsync(dinv, 0, N_NODES * sizeof(float), stream);
  k_deg<<<dim3(E_EDGES / 256), 256, 0, stream>>>(edge, dinv);
  k_dinv<<<dim3((N_NODES + 255) / 256), 256, 0, stream>>>(dinv);

  // --- encoder GCN ---
  k_enc_mm<<<dim3(N_NODES / 16, T_STEPS), 256, 0, stream>>>(x, enc_W, bufA);
  (void)hipMemsetAsync(bufB, 0, TNH * sizeof(float), stream);
  k_gcn_agg<<<dim3(E_EDGES), 128, 0, stream>>>(bufA, edge, dinv, bufB, HID);
  k_ln_enc<<<dim3(N_NODES, T_STEPS), 128, 0, stream>>>(bufB, bufA, dinv,
                                                       enc_b, enc_ln_g, enc_ln_b);

  // --- LSTM1 (time recurrence, parallel over nodes; TDM weight staging) ---
  size_t lds1 = (size_t)(128 * 256 + 64 * 256 + 16 * 128 + 16 * 64 + 16 * 256) * 4;
  k_lstm1<<<dim3(N_NODES / 16), 512, lds1, stream>>>(bufB, l1_Wih, l1_Whh, l1_b, zout);

  // --- LSTM2 (serial over N, persistent single workgroup) ---
  size_t lds2 = (size_t)(1024 + 2048 + 8192) * 4;
  k_lstm2<<<dim3(1), 1024, lds2, stream>>>(zout, l2_Wih, l2_Whh, l2_b, bufB);

  // --- decoder LN + GCN ---
  k_ln_dec<<<dim3(N_NODES, T_STEPS), 128, 0, stream>>>(bufB, dec_ln_g, dec_ln_b);
  k_dec_mm<<<dim3(N_NODES / 16, T_STEPS), 128, 0, stream>>>(bufB, dec_W, bufA);
  (void)hipMemsetAsync(bufA + TNF, 0, TNF * sizeof(float), stream);
  k_gcn_agg<<<dim3(E_EDGES), 64, 0, stream>>>(bufA, edge, dinv, bufA + TNF, F_DIM);

  // --- assemble reconstructed output ---
  k_final<<<dim3(N_NODES, T_STEPS), 65, 0, stream>>>(bufA + TNF, bufA, dinv,
                                                     dec_b, x, rec);
}